// BalanceL1Loss_86517821211947
// MI455X (gfx1250) — compile-verified
//
#include <hip/hip_runtime.h>
#include <hip/hip_bf16.h>
#include <stdint.h>

// ---------------------------------------------------------------------------
// BalanceL1Loss on gfx1250: exact radix-select top-k (no sort) + streaming
// reductions. WMMA f32 16x16x4 (A = ones) used as a free cross-lane adder.
// All histograms LDS-privatized (ds_add_u32) and flushed with integer global
// atomics (order-independent => deterministic across graph replays).
// ---------------------------------------------------------------------------

#define BL_NB 1024            // max blocks for streaming passes
#define BL_NT 256             // threads per block (8 wave32 waves)
#define BL_NWAVES (BL_NT / 32)

typedef __attribute__((ext_vector_type(2))) float v2f;
typedef __attribute__((ext_vector_type(8))) float v8f;

struct BL_Ctl {
    double sum_mask, sum_pos, kval, pos_cnt;
    unsigned b1, r1, b2, r2, tkey, rt;
};

// D = ones(16x4) * B(4x16) + C : every B element lands in exactly one column
// sum, and all 16 rows of D are identical, so D[0] (VGPR0) per lane holds a
// column partial; 16-lane xor-reduction of D[0] yields the grand total.
__device__ __forceinline__ v8f bl_wmma_acc(v2f b, v8f c) {
    v2f ones;
    ones.x = 1.0f;
    ones.y = 1.0f;
    return __builtin_amdgcn_wmma_f32_16x16x4_f32(
        /*neg_a=*/false, ones, /*neg_b=*/false, b,
        /*c_mod=*/(short)0, c, /*reuse_a=*/false, /*reuse_b=*/false);
}

__device__ __forceinline__ float bl_wave_total(v8f c) {
    float w = c[0];
    // xor masks 1,2,4,8: reduce within each 16-lane half (halves are duplicates)
    w += __shfl_xor(w, 1);
    w += __shfl_xor(w, 2);
    w += __shfl_xor(w, 4);
    w += __shfl_xor(w, 8);
    return w;
}

// ---------------------------------------------------------------------------
__global__ void bl_init(unsigned* h1, unsigned* h2, unsigned* h3) {
    for (int i = threadIdx.x; i < 4096; i += BL_NT) { h1[i] = 0u; h2[i] = 0u; }
    for (int i = threadIdx.x; i < 256;  i += BL_NT) { h3[i] = 0u; }
}

// Pass 1: loss, keys, level-1 histogram (key[31:20]), block sums of mask/pos.
__global__ void bl_pass1(const float* __restrict__ pred,
                         const float* __restrict__ gt,
                         const float* __restrict__ mask,
                         unsigned* __restrict__ keys,
                         unsigned* __restrict__ hist1,
                         double* __restrict__ part1,
                         unsigned ngroups, unsigned n) {
    __shared__ unsigned lhist[4096];
    __shared__ double wsum[2 * BL_NWAVES];
    for (int i = threadIdx.x; i < 4096; i += BL_NT) lhist[i] = 0u;
    __syncthreads();

    const float4* p4 = (const float4*)pred;
    const float4* g4 = (const float4*)gt;
    const float4* m4 = (const float4*)mask;
    uint4* k4 = (uint4*)keys;

    unsigned stride = gridDim.x * BL_NT;
    unsigned t0 = blockIdx.x * BL_NT + threadIdx.x;
    unsigned iters = (ngroups + stride - 1u) / stride;   // uniform trip count

    v8f cpos = {};
    v8f cmsk = {};

    for (unsigned it = 0; it < iters; ++it) {
        unsigned idx = t0 + it * stride;
        bool valid = idx < ngroups;
        unsigned gi = valid ? idx : 0u;              // clamped (safe) load
        float4 p = p4[gi];
        float4 g = g4[gi];
        float4 m = m4[gi];
        float l0 = fabsf(p.x - g.x), l1 = fabsf(p.y - g.y);
        float l2 = fabsf(p.z - g.z), l3 = fabsf(p.w - g.w);
        float q = valid ? 1.0f : 0.0f;
        float m0 = m.x * q, m1 = m.y * q, m2 = m.z * q, m3 = m.w * q;
        float pos0 = l0 * m0, pos1 = l1 * m1, pos2 = l2 * m2, pos3 = l3 * m3;

        if (valid) {
            uint4 kk;
            kk.x = __float_as_uint(l0 * (1.0f - m.x));
            kk.y = __float_as_uint(l1 * (1.0f - m.y));
            kk.z = __float_as_uint(l2 * (1.0f - m.z));
            kk.w = __float_as_uint(l3 * (1.0f - m.w));
            k4[idx] = kk;
            atomicAdd(&lhist[kk.x >> 20], 1u);
            atomicAdd(&lhist[kk.y >> 20], 1u);
            atomicAdd(&lhist[kk.z >> 20], 1u);
            atomicAdd(&lhist[kk.w >> 20], 1u);
        }
        // re-converged here: EXEC all-ones for WMMA
        v2f bp; bp.x = pos0 + pos1; bp.y = pos2 + pos3;
        v2f bm; bm.x = m0 + m1;     bm.y = m2 + m3;
        cpos = bl_wmma_acc(bp, cpos);
        cmsk = bl_wmma_acc(bm, cmsk);
    }

    float wp = bl_wave_total(cpos);
    float wm = bl_wave_total(cmsk);
    unsigned lane = threadIdx.x & 31u, wid = threadIdx.x >> 5;
    if (lane == 0) { wsum[wid] = (double)wp; wsum[BL_NWAVES + wid] = (double)wm; }
    __syncthreads();

    for (int i = threadIdx.x; i < 4096; i += BL_NT) {
        unsigned c = lhist[i];
        if (c) atomicAdd(&hist1[i], c);
    }

    if (threadIdx.x == 0) {
        double sp = 0.0, sm = 0.0;
        for (int w = 0; w < BL_NWAVES; ++w) { sp += wsum[w]; sm += wsum[BL_NWAVES + w]; }
        if (blockIdx.x == 0) {   // scalar tail (n % 4 elements), once
            for (unsigned i = ngroups * 4u; i < n; ++i) {
                float l = fabsf(pred[i] - gt[i]);
                float mm = mask[i];
                sp += (double)(l * mm);
                sm += (double)mm;
                unsigned key = __float_as_uint(l * (1.0f - mm));
                keys[i] = key;
                atomicAdd(&hist1[key >> 20], 1u);
            }
        }
        part1[2u * blockIdx.x]      = sm;
        part1[2u * blockIdx.x + 1u] = sp;
    }
}

// Serial combine + level-1 scan (deterministic, once per launch).
__global__ void bl_scan1(const unsigned* __restrict__ hist1,
                         const double* __restrict__ part1,
                         BL_Ctl* ctl, unsigned nb, unsigned n) {
    if (threadIdx.x != 0) return;
    double sm = 0.0, sp = 0.0;
    for (unsigned b = 0; b < nb; ++b) { sm += part1[2u * b]; sp += part1[2u * b + 1u]; }
    double posc  = floor(sm);
    double negav = floor((double)n - sm);
    double kd    = fmin(negav, posc * 3.0);
    ctl->sum_mask = sm; ctl->sum_pos = sp; ctl->pos_cnt = posc; ctl->kval = kd;
    unsigned kk = (unsigned)kd;          // integer-valued by construction
    unsigned cum = 0, b1 = 0, r1 = 0;
    for (int b = 4095; b >= 0; --b) {
        unsigned c = hist1[b];
        if (cum + c >= kk) { b1 = (unsigned)b; r1 = kk - cum; break; }
        cum += c;
    }
    ctl->b1 = b1; ctl->r1 = r1;
}

// Level-2 histogram: key[19:8] among keys whose key[31:20] == b1.
// LDS-privatized: hot-bin atomic traffic stays in the WGP.
__global__ void bl_pass2(const unsigned* __restrict__ keys,
                         unsigned* __restrict__ hist2,
                         const BL_Ctl* __restrict__ ctl,
                         unsigned ngroups, unsigned n) {
    __shared__ unsigned lhist[4096];
    for (int i = threadIdx.x; i < 4096; i += BL_NT) lhist[i] = 0u;
    __syncthreads();

    unsigned b1 = ctl->b1;
    const uint4* k4 = (const uint4*)keys;
    unsigned stride = gridDim.x * BL_NT;
    unsigned t0 = blockIdx.x * BL_NT + threadIdx.x;
    for (unsigned idx = t0; idx < ngroups; idx += stride) {
        uint4 k = k4[idx];
        if ((k.x >> 20) == b1) atomicAdd(&lhist[(k.x >> 8) & 0xFFFu], 1u);
        if ((k.y >> 20) == b1) atomicAdd(&lhist[(k.y >> 8) & 0xFFFu], 1u);
        if ((k.z >> 20) == b1) atomicAdd(&lhist[(k.z >> 8) & 0xFFFu], 1u);
        if ((k.w >> 20) == b1) atomicAdd(&lhist[(k.w >> 8) & 0xFFFu], 1u);
    }
    if (t0 == 0) {
        for (unsigned i = ngroups * 4u; i < n; ++i) {
            unsigned k = keys[i];
            if ((k >> 20) == b1) atomicAdd(&lhist[(k >> 8) & 0xFFFu], 1u);
        }
    }
    __syncthreads();
    for (int i = threadIdx.x; i < 4096; i += BL_NT) {
        unsigned c = lhist[i];
        if (c) atomicAdd(&hist2[i], c);
    }
}

__global__ void bl_scan2(const unsigned* __restrict__ hist2, BL_Ctl* ctl) {
    if (threadIdx.x != 0) return;
    unsigned target = ctl->r1, cum = 0, b2 = 0, r2 = 0;
    for (int b = 4095; b >= 0; --b) {
        unsigned c = hist2[b];
        if (cum + c >= target) { b2 = (unsigned)b; r2 = target - cum; break; }
        cum += c;
    }
    ctl->b2 = b2; ctl->r2 = r2;
}

// Level-3 histogram: key[7:0] among keys whose key[31:8] == (b1<<12)|b2.
__global__ void bl_pass2b(const unsigned* __restrict__ keys,
                          unsigned* __restrict__ hist3,
                          const BL_Ctl* __restrict__ ctl,
                          unsigned ngroups, unsigned n) {
    __shared__ unsigned lhist[256];
    if (threadIdx.x < 256) lhist[threadIdx.x] = 0u;
    __syncthreads();

    unsigned pref = (ctl->b1 << 12) | ctl->b2;
    const uint4* k4 = (const uint4*)keys;
    unsigned stride = gridDim.x * BL_NT;
    unsigned t0 = blockIdx.x * BL_NT + threadIdx.x;
    for (unsigned idx = t0; idx < ngroups; idx += stride) {
        uint4 k = k4[idx];
        if ((k.x >> 8) == pref) atomicAdd(&lhist[k.x & 0xFFu], 1u);
        if ((k.y >> 8) == pref) atomicAdd(&lhist[k.y & 0xFFu], 1u);
        if ((k.z >> 8) == pref) atomicAdd(&lhist[k.z & 0xFFu], 1u);
        if ((k.w >> 8) == pref) atomicAdd(&lhist[k.w & 0xFFu], 1u);
    }
    if (t0 == 0) {
        for (unsigned i = ngroups * 4u; i < n; ++i) {
            unsigned k = keys[i];
            if ((k >> 8) == pref) atomicAdd(&lhist[k & 0xFFu], 1u);
        }
    }
    __syncthreads();
    if (threadIdx.x < 256) {
        unsigned c = lhist[threadIdx.x];
        if (c) atomicAdd(&hist3[threadIdx.x], c);
    }
}

__global__ void bl_scan3(const unsigned* __restrict__ hist3, BL_Ctl* ctl) {
    if (threadIdx.x != 0) return;
    unsigned target = ctl->r2, cum = 0, b3 = 255, rt = 0;
    for (int b = 255; b >= 0; --b) {
        unsigned c = hist3[b];
        if (cum + c >= target) { b3 = (unsigned)b; rt = target - cum; break; }
        cum += c;
    }
    ctl->tkey = (ctl->b1 << 20) | (ctl->b2 << 8) | b3;
    ctl->rt = rt;
}

// Pass 3: sum of values with key strictly above threshold T.
__global__ void bl_pass3(const unsigned* __restrict__ keys,
                         const BL_Ctl* __restrict__ ctl,
                         double* __restrict__ part7,
                         unsigned ngroups, unsigned n) {
    __shared__ double wsum[BL_NWAVES];
    unsigned T = ctl->tkey;
    const uint4* k4 = (const uint4*)keys;
    unsigned stride = gridDim.x * BL_NT;
    unsigned t0 = blockIdx.x * BL_NT + threadIdx.x;
    unsigned iters = (ngroups + stride - 1u) / stride;   // uniform trip count

    v8f acc = {};
    for (unsigned it = 0; it < iters; ++it) {
        unsigned idx = t0 + it * stride;
        bool valid = idx < ngroups;
        uint4 k = k4[valid ? idx : 0u];
        float v0 = (valid && k.x > T) ? __uint_as_float(k.x) : 0.0f;
        float v1 = (valid && k.y > T) ? __uint_as_float(k.y) : 0.0f;
        float v2 = (valid && k.z > T) ? __uint_as_float(k.z) : 0.0f;
        float v3 = (valid && k.w > T) ? __uint_as_float(k.w) : 0.0f;
        v2f b; b.x = v0 + v1; b.y = v2 + v3;
        acc = bl_wmma_acc(b, acc);
    }

    float w = bl_wave_total(acc);
    unsigned lane = threadIdx.x & 31u, wid = threadIdx.x >> 5;
    if (lane == 0) wsum[wid] = (double)w;
    __syncthreads();
    if (threadIdx.x == 0) {
        double s = 0.0;
        for (int ww = 0; ww < BL_NWAVES; ++ww) s += wsum[ww];
        if (blockIdx.x == 0) {
            for (unsigned i = ngroups * 4u; i < n; ++i) {
                unsigned kk = keys[i];
                if (kk > T) s += (double)__uint_as_float(kk);
            }
        }
        part7[blockIdx.x] = s;
    }
}

__global__ void bl_final(const double* __restrict__ part7,
                         const BL_Ctl* __restrict__ ctl,
                         float* __restrict__ out, unsigned nb) {
    if (threadIdx.x != 0) return;
    double ns = 0.0;
    for (unsigned b = 0; b < nb; ++b) ns += part7[b];
    ns += (double)ctl->rt * (double)__uint_as_float(ctl->tkey);
    double pl = ctl->sum_pos / ctl->pos_cnt;   // div-by-zero matches reference
    double nl = ns / ctl->kval;
    out[0] = (float)(pl + nl);
    out[1] = (float)pl;
    out[2] = (float)nl;
}

// ---------------------------------------------------------------------------
extern "C" void kernel_launch(void* const* d_in, const int* in_sizes, int n_in,
                              void* d_out, int out_size, void* d_ws, size_t ws_size,
                              hipStream_t stream) {
    const float* pred = (const float*)d_in[0];
    const float* gt   = (const float*)d_in[1];
    const float* mask = (const float*)d_in[2];
    unsigned n = (unsigned)in_sizes[1];        // N*H*W
    if (n == 0) return;
    unsigned ngroups = n / 4u;

    // Workspace layout
    char* ws = (char*)d_ws;
    size_t off = 0;
    unsigned* keys = (unsigned*)(ws + off); off += (size_t)n * 4u;
    off = (off + 255u) & ~(size_t)255u;
    BL_Ctl* ctl = (BL_Ctl*)(ws + off);      off += 256u;
    unsigned* h1 = (unsigned*)(ws + off);   off += 4096u * 4u;
    unsigned* h2 = (unsigned*)(ws + off);   off += 4096u * 4u;
    unsigned* h3 = (unsigned*)(ws + off);   off += 256u * 4u;
    off = (off + 255u) & ~(size_t)255u;
    double* part1 = (double*)(ws + off);    off += (size_t)BL_NB * 2u * 8u;
    double* part7 = (double*)(ws + off);    off += (size_t)BL_NB * 8u;

    unsigned nb = (ngroups + BL_NT - 1u) / BL_NT;
    if (nb > BL_NB) nb = BL_NB;
    if (nb == 0) nb = 1;

    bl_init  <<<1, BL_NT, 0, stream>>>(h1, h2, h3);
    bl_pass1 <<<nb, BL_NT, 0, stream>>>(pred, gt, mask, keys, h1, part1, ngroups, n);
    bl_scan1 <<<1, 64, 0, stream>>>(h1, part1, ctl, nb, n);
    bl_pass2 <<<nb, BL_NT, 0, stream>>>(keys, h2, ctl, ngroups, n);
    bl_scan2 <<<1, 64, 0, stream>>>(h2, ctl);
    bl_pass2b<<<nb, BL_NT, 0, stream>>>(keys, h3, ctl, ngroups, n);
    bl_scan3 <<<1, 64, 0, stream>>>(h3, ctl);
    bl_pass3 <<<nb, BL_NT, 0, stream>>>(keys, ctl, part7, ngroups, n);
    bl_final <<<1, 64, 0, stream>>>(part7, ctl, (float*)d_out, nb);
}